// EKFGradient_55568286875828
// MI455X (gfx1250) — compile-verified
//
#include <hip/hip_runtime.h>

typedef __attribute__((ext_vector_type(2))) float v2f;
typedef __attribute__((ext_vector_type(8))) float v8f;

#define DT_       (1.0f/120.0f)
#define NROWS     16384          // raw_data rows
#define NSTEPS    16381          // N
#define JMAX      16382          // data rows - 1
#define TWOPI_F   6.28318530717958647692f
#define INV2PI_F  0.15915494309189533577f

__device__ __forceinline__ float sel6(int n, float a0, float a1, float a2,
                                      float a3, float a4, float a5) {
    float v = 0.0f;
    v = (n == 0) ? a0 : v;
    v = (n == 1) ? a1 : v;
    v = (n == 2) ? a2 : v;
    v = (n == 3) ? a3 : v;
    v = (n == 4) ? a4 : v;
    v = (n == 5) ? a5 : v;
    return v;
}

// Branch-free tanh: no EXEC save/restore on the serial critical path.
// tanh(x) = sign(x) * (1 - 2/(exp(2|x|)+1)); exp->inf saturates to 1 correctly.
__device__ __forceinline__ float tanh_fast(float x) {
    const float a = __builtin_fabsf(x);
    const float e = __expf(2.0f * a);                       // v_exp_f32 path
    const float t = 1.0f - 2.0f * __builtin_amdgcn_rcpf(e + 1.0f);
    return __builtin_copysignf(t, x);
}

__global__ void __launch_bounds__(256, 1)
ekf_loss_kernel(const float* __restrict__ raw,
                const float* __restrict__ state,
                const float* __restrict__ dyna,
                float* __restrict__ out)
{
    extern __shared__ float smem[];
    float* rawl = smem;              // 65536 floats: full raw_data (16384 x 4)
    float* Pl   = smem + NROWS * 4;  // 256 floats: P as 16x16 row-major (6x6 live)

    const int tid = threadIdx.x;

    // ---- Stage all of raw_data into LDS (8 waves cooperatively) ----
    {
        const float4* g4 = (const float4*)raw;
        float4*       l4 = (float4*)rawl;
        for (int r = tid; r < NROWS; r += 256) l4[r] = g4[r];
    }
    __syncthreads();
    if (tid >= 32) return;           // one wave runs the sequential scan

    const int lane = tid;
    const int half = lane >> 4;      // 0: lanes 0-15, 1: lanes 16-31
    const int n    = lane & 15;

    // ---- P0 = 0.01 * I(6) embedded in zeroed 16x16 ----
    #pragma unroll
    for (int k = 0; k < 8; ++k) {
        int idx = lane * 8 + k;
        int rr = idx >> 4, cc = idx & 15;
        Pl[idx] = (rr == cc && rr < 6) ? 0.01f : 0.0f;
    }
    __threadfence_block();

    const float friction = dyna[0];
    const float damping  = dyna[1];
    float px = state[0], py = state[1], vx = state[2],
          vy = state[3], th = state[4], om = state[5];

    const float t0 = rawl[1 * 4 + 3];   // t[0] = raw[1].time
    const float t1 = rawl[2 * 4 + 3];   // t[1] = raw[2].time

    // ---- Q diagonal in WMMA C/D layout (vgpr r <-> row r+8*half, col n) ----
    v8f qc = {0.f, 0.f, 0.f, 0.f, 0.f, 0.f, 0.f, 0.f};
    qc[0] = (half == 0 && n == 0) ? 2e-10f : 0.0f;
    qc[1] = (half == 0 && n == 1) ? 2e-10f : 0.0f;
    qc[2] = (half == 0 && n == 2) ? 3e-7f  : 0.0f;
    qc[3] = (half == 0 && n == 3) ? 3e-7f  : 0.0f;
    qc[4] = (half == 0 && n == 4) ? 0.01f  : 0.0f;
    qc[5] = (half == 0 && n == 5) ? 0.1f   : 0.0f;

    // ---- Per-lane templates for F in A-layout (== F^T in B-layout) ----
    // chunk0 covers K=0..3 (half0 -> K0,K1 ; half1 -> K2,K3); chunk1 covers K=4..7
    const float af0x_base = (half == 0) ? ((n == 0) ? 1.0f : 0.0f)
                                        : ((n == 0) ? DT_  : 0.0f);
    const float af0y_base = (half == 0) ? ((n == 1) ? 1.0f : 0.0f)
                                        : ((n == 1) ? DT_  : 0.0f);
    const bool isf22 = (half == 1) && (n == 2);   // F[2][2] slot
    const bool isf33 = (half == 1) && (n == 3);   // F[3][3] slot
    v2f af1;
    af1.x = (half == 0 && n == 4) ? 1.0f : 0.0f;
    af1.y = (half == 0) ? ((n == 4) ? DT_ : ((n == 5) ? 1.0f : 0.0f)) : 0.0f;

    // LDS float-index bases: A-operand of P (row n, cols kb..kb+1), P store
    const int apbase0 = n * 16 + 2 * half;       // chunk0
    const int apbase1 = apbase0 + 4;             // chunk1
    const int pstbase = 128 * half + n;          // + r*16 -> (r+8*half, n)

    const v8f zero8 = {0.f, 0.f, 0.f, 0.f, 0.f, 0.f, 0.f, 0.f};

    int   j = 1, num = 0;
    float evsum = 0.0f;

    for (int i = 1; i <= NSTEPS; ++i) {
        const float fi = (float)i;

        // ---- measurement gating (uniform) ----
        const int jp = j + 1;
        const int jc = (j  < 0) ? 0 : ((j  > JMAX) ? JMAX : j );
        const int jn = (jp < 0) ? 0 : ((jp > JMAX) ? JMAX : jp);
        const float tj = rawl[(jc + 1) * 4 + 3] - t0;
        const float z0 = rawl[(jn + 1) * 4 + 0];
        const float z1 = rawl[(jn + 1) * 4 + 1];
        const float z2 = rawl[(jn + 1) * 4 + 2];
        const float tn = rawl[(jn + 1) * 4 + 3];
        const float lo = (fi - 0.2f) * DT_;
        const float hi = (fi + 0.2f) * DT_;
        const bool cond1 = (lo < tj) && (tj < hi);
        const bool cond2 = (tn - t1) <= lo;

        // ---- state propagation + analytic Jacobian entries ----
        const float tvx = tanh_fast(vx * 100.0f);
        const float tvy = tanh_fast(vy * 100.0f);
        const float ax  = -(damping * vx + friction * tvx);
        const float ay  = -(damping * vy + friction * tvy);
        const float xp0 = px + DT_ * vx;
        const float xp1 = py + DT_ * vy;
        const float xp2 = vx + DT_ * ax;
        const float xp3 = vy + DT_ * ay;
        const float xp4 = th + DT_ * om;
        const float xp5 = om;
        const float f22 = 1.0f - DT_ * (damping + friction * 100.0f * (1.0f - tvx * tvx));
        const float f33 = 1.0f - DT_ * (damping + friction * 100.0f * (1.0f - tvy * tvy));

        v2f af0;
        af0.x = isf22 ? f22 : af0x_base;
        af0.y = isf33 ? f33 : af0y_base;

        // ---- G = P * F^T : A = P from LDS (contiguous), B = F^T (== af regs) ----
        const v2f ap0 = *(const v2f*)&Pl[apbase0];
        const v2f ap1 = *(const v2f*)&Pl[apbase1];
        v8f g2 = __builtin_amdgcn_wmma_f32_16x16x4_f32(false, ap0, false, af0,
                                                       (short)0, zero8, false, false);
        g2     = __builtin_amdgcn_wmma_f32_16x16x4_f32(false, ap1, false, af1,
                                                       (short)0, g2,    false, false);

        // ---- C/D layout of G -> B layout via lane-16 swap ----
        v2f bg0, bg1;
        bg0.x = half ? __shfl_xor(g2[2], 16, 32) : g2[0];
        bg0.y = half ? __shfl_xor(g2[3], 16, 32) : g2[1];
        bg1.x = half ? __shfl_xor(g2[6], 16, 32) : g2[4];
        bg1.y = half ? __shfl_xor(g2[7], 16, 32) : g2[5];

        // ---- Pp = F * G + Q ----
        v8f pp = __builtin_amdgcn_wmma_f32_16x16x4_f32(false, af0, false, bg0,
                                                       (short)0, qc, false, false);
        pp     = __builtin_amdgcn_wmma_f32_16x16x4_f32(false, af1, false, bg1,
                                                       (short)0, pp, false, false);

        // ---- broadcast Pp[m][{0,1,4}] (rows m live in vgpr m, lanes 0-15) ----
        const float P00 = __shfl(pp[0], 0, 32), P01 = __shfl(pp[0], 1, 32), P04 = __shfl(pp[0], 4, 32);
        const float P10 = __shfl(pp[1], 0, 32), P11 = __shfl(pp[1], 1, 32), P14 = __shfl(pp[1], 4, 32);
        const float P20 = __shfl(pp[2], 0, 32), P21 = __shfl(pp[2], 1, 32), P24 = __shfl(pp[2], 4, 32);
        const float P30 = __shfl(pp[3], 0, 32), P31 = __shfl(pp[3], 1, 32), P34 = __shfl(pp[3], 4, 32);
        const float P40 = __shfl(pp[4], 0, 32), P41 = __shfl(pp[4], 1, 32), P44 = __shfl(pp[4], 4, 32);
        const float P50 = __shfl(pp[5], 0, 32), P51 = __shfl(pp[5], 1, 32), P54 = __shfl(pp[5], 4, 32);

        // ---- S = H Pp H^T + R, 3x3 inverse by adjugate ----
        const float s00 = P00 + 2.5e-7f, s01 = P01,           s02 = P04;
        const float s10 = P10,           s11 = P11 + 2.5e-7f, s12 = P14;
        const float s20 = P40,           s21 = P41,           s22 = P44 + 0.0091f;
        const float c00 = s11 * s22 - s12 * s21;
        const float c01 = s12 * s20 - s10 * s22;
        const float c02 = s10 * s21 - s11 * s20;
        const float det = s00 * c00 + s01 * c01 + s02 * c02;
        const float rdet = __builtin_amdgcn_rcpf(det);       // v_rcp_f32, branch-free
        const float Si00 = c00 * rdet, Si01 = (s02 * s21 - s01 * s22) * rdet, Si02 = (s01 * s12 - s02 * s11) * rdet;
        const float Si10 = c01 * rdet, Si11 = (s00 * s22 - s02 * s20) * rdet, Si12 = (s02 * s10 - s00 * s12) * rdet;
        const float Si20 = c02 * rdet, Si21 = (s01 * s20 - s00 * s21) * rdet, Si22 = (s00 * s11 - s01 * s10) * rdet;

        // ---- innovation (angle wrapped to (-pi, pi]) ----
        const float y0 = z0 - xp0;
        const float y1 = z1 - xp1;
        const float yr = z2 - xp4;
        const float y2 = yr - TWOPI_F * rintf(yr * INV2PI_F);

        const float g = cond1 ? 1.0f : 0.0f;

        // ---- K = Pp[:,{0,1,4}] * Sinv, gated by cond1 ----
        const float K00 = (P00*Si00 + P01*Si10 + P04*Si20)*g, K01 = (P00*Si01 + P01*Si11 + P04*Si21)*g, K02 = (P00*Si02 + P01*Si12 + P04*Si22)*g;
        const float K10 = (P10*Si00 + P11*Si10 + P14*Si20)*g, K11 = (P10*Si01 + P11*Si11 + P14*Si21)*g, K12 = (P10*Si02 + P11*Si12 + P14*Si22)*g;
        const float K20 = (P20*Si00 + P21*Si10 + P24*Si20)*g, K21 = (P20*Si01 + P21*Si11 + P24*Si21)*g, K22 = (P20*Si02 + P21*Si12 + P24*Si22)*g;
        const float K30 = (P30*Si00 + P31*Si10 + P34*Si20)*g, K31 = (P30*Si01 + P31*Si11 + P34*Si21)*g, K32 = (P30*Si02 + P31*Si12 + P34*Si22)*g;
        const float K40 = (P40*Si00 + P41*Si10 + P44*Si20)*g, K41 = (P40*Si01 + P41*Si11 + P44*Si21)*g, K42 = (P40*Si02 + P41*Si12 + P44*Si22)*g;
        const float K50 = (P50*Si00 + P51*Si10 + P54*Si20)*g, K51 = (P50*Si01 + P51*Si11 + P54*Si21)*g, K52 = (P50*Si02 + P51*Si12 + P54*Si22)*g;

        // ---- ev = det(S) + y' Sinv y, gated ----
        const float Sy0 = Si00 * y0 + Si01 * y1 + Si02 * y2;
        const float Sy1 = Si10 * y0 + Si11 * y1 + Si12 * y2;
        const float Sy2 = Si20 * y0 + Si21 * y1 + Si22 * y2;
        evsum += g * (det + y0 * Sy0 + y1 * Sy1 + y2 * Sy2);

        // ---- x update (K already gated -> identity when cond1 false) ----
        px = xp0 + (K00 * y0 + K01 * y1 + K02 * y2);
        py = xp1 + (K10 * y0 + K11 * y1 + K12 * y2);
        vx = xp2 + (K20 * y0 + K21 * y1 + K22 * y2);
        vy = xp3 + (K30 * y0 + K31 * y1 + K32 * y2);
        th = xp4 + (K40 * y0 + K41 * y1 + K42 * y2);
        om = xp5 + (K50 * y0 + K51 * y1 + K52 * y2);

        // ---- P' = Pp + (-K_eff) * (H Pp) : one WMMA, K-dim 3 padded to 4 ----
        v2f ak;
        const float kc0 = sel6(n, K00, K10, K20, K30, K40, K50);
        const float kc1 = sel6(n, K01, K11, K21, K31, K41, K51);
        const float kc2 = sel6(n, K02, K12, K22, K32, K42, K52);
        ak.x = half ? -kc2 : -kc0;
        ak.y = half ? 0.0f : -kc1;
        v2f bh;
        bh.x = half ? __shfl_xor(pp[4], 16, 32) : pp[0];   // rows {0,1,4,0pad}
        bh.y = half ? 0.0f : pp[1];
        const v8f pn = __builtin_amdgcn_wmma_f32_16x16x4_f32(false, ak, false, bh,
                                                             (short)0, pp, false, false);

        // ---- write P back to LDS for next iteration ----
        #pragma unroll
        for (int r = 0; r < 8; ++r) Pl[pstbase + r * 16] = pn[r];
        __threadfence_block();

        j   += (cond1 || cond2) ? 1 : 0;
        num += cond1 ? 1 : 0;
    }

    if (lane == 0) {
        const float dn = (float)((num > 1) ? num : 1);
        out[0] = evsum / dn;
    }
}

extern "C" void kernel_launch(void* const* d_in, const int* in_sizes, int n_in,
                              void* d_out, int out_size, void* d_ws, size_t ws_size,
                              hipStream_t stream) {
    (void)in_sizes; (void)n_in; (void)out_size; (void)d_ws; (void)ws_size;
    const float* raw   = (const float*)d_in[0];
    const float* state = (const float*)d_in[1];
    const float* dyna  = (const float*)d_in[2];
    float*       out   = (float*)d_out;

    const size_t shmem = (size_t)(NROWS * 4 + 256) * sizeof(float); // 263,168 B < 320 KB WGP LDS
    hipFuncSetAttribute(reinterpret_cast<const void*>(&ekf_loss_kernel),
                        hipFuncAttributeMaxDynamicSharedMemorySize, (int)shmem);
    ekf_loss_kernel<<<dim3(1), dim3(256), shmem, stream>>>(raw, state, dyna, out);
}